// PCEN_59081570125217
// MI455X (gfx1250) — compile-verified
//
#include <hip/hip_runtime.h>
#include <math.h>

// ---------------- problem constants ----------------
#define BB      64
#define CC      128
#define TT      4000
#define ROWS    (BB * CC)        // 8192 independent EMA rows
#define NCHUNK  8
#define LCH     (TT / NCHUNK)    // 500 timesteps per chunk
#define TW      20               // tile width (floats) staged through LDS
#define NTILE   (LCH / TW)       // 25 tiles per chunk
#define BLK     256              // threads per block (= rows per tile)

#define S_          0.025f
#define A_          0.975f
#define ALPHA_      0.98f
#define EPS_        1e-6f
#define SQRT_DELTA  1.41421356237309515f   // sqrt(2.0)

// ---------------- CDNA5 async global->LDS path ----------------
#if defined(__has_builtin)
#if __has_builtin(__builtin_amdgcn_global_load_async_to_lds_b128) && \
    __has_builtin(__builtin_amdgcn_s_wait_asynccnt)
#define USE_ASYNC_LDS 1
#endif
#if __has_builtin(__builtin_amdgcn_exp2f) && __has_builtin(__builtin_amdgcn_logf)
#define FAST_EXP2(v) __builtin_amdgcn_exp2f(v)   // v_exp_f32
#define FAST_LOG2(v) __builtin_amdgcn_logf(v)    // v_log_f32 (base-2)
#endif
#endif
#ifndef USE_ASYNC_LDS
#define USE_ASYNC_LDS 0
#endif
#ifndef FAST_EXP2
#define FAST_EXP2(v) exp2f(v)
#define FAST_LOG2(v) log2f(v)
#endif

#define AS1 __attribute__((address_space(1)))
#define AS3 __attribute__((address_space(3)))
typedef int v4i __attribute__((ext_vector_type(4)));

// =====================================================================
// MAIN PASS (placed first in the module so the disasm snippet shows it).
// Block = 256 rows x 1 chunk. Input staged through LDS tiles [256 x 20]
// (double buffered) via async global->LDS b128 copies pipelined on
// ASYNCcnt; each thread then walks its own row from LDS, computing
//   M_t = (1-s) M_{t-1} + s x_t
//   y_t = sqrt(x_t * (eps+M_t)^-0.98 + 2) - sqrt(2)
// starting from the exact per-chunk M computed by the two helper kernels.
// =====================================================================
__global__ __launch_bounds__(BLK)
void pcen_main(const float* __restrict__ x, const float* __restrict__ Ms,
               float* __restrict__ y) {
    __shared__ float tile[2][BLK * TW];               // 2 x 20 KB

    const int tid     = threadIdx.x;
    const int rowBase = blockIdx.x * BLK;
    const int ch      = blockIdx.y;

    // cooperative tile fetch: 1280 float4s, lanes hit consecutive addresses
    auto issue = [&](int tl, int buf) {
#pragma unroll
        for (int k = 0; k < 5; ++k) {
            const int e4 = k * BLK + tid;             // 0..1279
            const int r  = e4 / 5;                    // tile row (0..255)
            const int c4 = e4 - r * 5;                // float4 col (0..4)
            const float* g = x + (size_t)(rowBase + r) * TT
                               + (size_t)ch * LCH + tl * TW + c4 * 4;
            float* l = &tile[buf][r * TW + c4 * 4];
#if USE_ASYNC_LDS
            __builtin_amdgcn_global_load_async_to_lds_b128(
                (AS1 v4i*)g, (AS3 v4i*)l, 0, 0);
#else
            *(float4*)l = *(const float4*)g;
#endif
        }
    };

    float m = Ms[(size_t)ch * ROWS + rowBase + tid];  // exact M entering chunk
    float* __restrict__ outp = y + (size_t)(rowBase + tid) * TT + (size_t)ch * LCH;

    issue(0, 0);
    for (int tl = 0; tl < NTILE; ++tl) {
        const int buf = tl & 1;
        if (tl + 1 < NTILE) {
            issue(tl + 1, buf ^ 1);                   // prefetch next tile
#if USE_ASYNC_LDS
            __builtin_amdgcn_s_wait_asynccnt(5);      // oldest 5 (this tile) done
#endif
        } else {
#if USE_ASYNC_LDS
            __builtin_amdgcn_s_wait_asynccnt(0);
#endif
        }
        __syncthreads();                              // all waves' data visible

        float yv[TW];
#pragma unroll
        for (int i = 0; i < TW; ++i) {
            const float xv = tile[buf][tid * TW + i];
            m = fmaf(A_, m, S_ * xv);                 // M_t = (1-s)M + s x
            const float p = m + EPS_;
            const float g = FAST_EXP2(-ALPHA_ * FAST_LOG2(p));  // (eps+M)^-0.98
            yv[i] = sqrtf(fmaf(xv, g, 2.0f)) - SQRT_DELTA;      // r = 0.5
        }
        float4* o4 = (float4*)(outp + tl * TW);
#pragma unroll
        for (int q = 0; q < 5; ++q)
            o4[q] = make_float4(yv[4 * q], yv[4 * q + 1],
                                yv[4 * q + 2], yv[4 * q + 3]);

        __syncthreads();                              // buffer reuse fence
    }
}

// =====================================================================
// Kernel 1: per-(row,chunk) EMA endpoint starting from M=0.
//   E_chunk = sum_i a^(L-1-i) * s * x_i   (linear part of the recurrence)
// 4-wide reassociation: m' = a^4 m + s*(a^3 x0 + a^2 x1 + a x2 + x3)
// =====================================================================
__global__ __launch_bounds__(BLK)
void pcen_endpoints(const float* __restrict__ x, float* __restrict__ E) {
    const int gid = blockIdx.x * BLK + threadIdx.x;   // 0 .. ROWS*NCHUNK-1
    const int row = gid & (ROWS - 1);                 // lanes -> consecutive rows
    const int ch  = gid >> 13;                        // gid / ROWS
    const float4* __restrict__ p =
        (const float4*)(x + (size_t)row * TT + (size_t)ch * LCH);

    const float A1 = A_;
    const float A2 = A_ * A_;
    const float A3 = A2 * A_;
    const float A4 = A2 * A2;

    float m = 0.0f;
#pragma unroll 5
    for (int i = 0; i < LCH / 4; ++i) {
        float4 v = p[i];
        float h = fmaf(A3, v.x, fmaf(A2, v.y, fmaf(A1, v.z, v.w)));
        m = fmaf(A4, m, S_ * h);
    }
    E[(size_t)ch * ROWS + row] = m;   // coalesced: [chunk][row]
}

// =====================================================================
// Kernel 2: tiny per-row scan over the 8 chunk endpoints.
//   Mstart[j] = M at end of chunk j-1 ;  M_j_end = a^L * M_{j-1}_end + E_j
// =====================================================================
__global__ __launch_bounds__(BLK)
void pcen_scan(const float* __restrict__ E, float* __restrict__ Ms, float aL) {
    const int row = blockIdx.x * BLK + threadIdx.x;
    float m = 0.0f;
#pragma unroll
    for (int j = 0; j < NCHUNK; ++j) {
        Ms[(size_t)j * ROWS + row] = m;
        m = fmaf(aL, m, E[(size_t)j * ROWS + row]);
    }
}

// =====================================================================
// Host launcher
// =====================================================================
extern "C" void kernel_launch(void* const* d_in, const int* in_sizes, int n_in,
                              void* d_out, int out_size, void* d_ws, size_t ws_size,
                              hipStream_t stream) {
    const float* x = (const float*)d_in[0];
    float* yout = (float*)d_out;

    float* E  = (float*)d_ws;                 // [NCHUNK][ROWS]  256 KB
    float* Ms = E + (size_t)ROWS * NCHUNK;    // [NCHUNK][ROWS]  256 KB

    const float aL = (float)pow((double)A_, (double)LCH);  // a^500, exact in double

    pcen_endpoints<<<(ROWS * NCHUNK) / BLK, BLK, 0, stream>>>(x, E);
    pcen_scan<<<ROWS / BLK, BLK, 0, stream>>>(E, Ms, aL);
    dim3 grid(ROWS / BLK, NCHUNK);
    pcen_main<<<grid, BLK, 0, stream>>>(x, Ms, yout);
}